// RegularLoop_4939212390750
// MI455X (gfx1250) — compile-verified
//
#include <hip/hip_runtime.h>

typedef float v2f __attribute__((ext_vector_type(2)));
typedef float v8f __attribute__((ext_vector_type(8)));

static constexpr int KD = 10;   // feature dim from reference

// ---------------------------------------------------------------------------
// Kernel 1: collapse the num_steps-step affine recurrence into (M, v):
//   A = W^T ; (M, v) <- (I, 0) ; repeat: (M, v) <- (M*A, v*A + (b+1))
// so that out = x @ M + v exactly (in exact arithmetic).
// Writes zero-padded 16x16 M (row-major, B-fragment source) to ws[0..255]
// and 16-wide padded v (C-fragment source) to ws[256..271].
// ---------------------------------------------------------------------------
__global__ __launch_bounds__(256) void precompute_affine(
    const float* __restrict__ W, const float* __restrict__ b,
    const int* __restrict__ steps_p, float* __restrict__ ws) {
  __shared__ float A[KD][KD];
  __shared__ float Mbuf[2][KD][KD];
  __shared__ float vbuf[2][KD];
  __shared__ float u[KD];
  const int tid = threadIdx.x;

  if (tid < KD * KD) {
    int k = tid / KD, n = tid % KD;
    A[k][n] = W[n * KD + k];                 // A = W^T  (out = x @ W^T)
    Mbuf[0][k][n] = (k == n) ? 1.0f : 0.0f;  // M = I
  }
  if (tid < KD) {
    u[tid] = b[tid] + 1.0f;
    vbuf[0][tid] = 0.0f;
  }
  __syncthreads();

  const int S = steps_p[0];
  int cur = 0;
  for (int t = 0; t < S; ++t) {
    int nxt = cur ^ 1;
    if (tid < KD * KD) {
      int i = tid / KD, j = tid % KD;
      float acc = 0.0f;
#pragma unroll
      for (int k = 0; k < KD; ++k) acc += Mbuf[cur][i][k] * A[k][j];
      Mbuf[nxt][i][j] = acc;
    }
    if (tid < KD) {
      float acc = u[tid];
#pragma unroll
      for (int k = 0; k < KD; ++k) acc += vbuf[cur][k] * A[k][tid];
      vbuf[nxt][tid] = acc;
    }
    __syncthreads();
    cur = nxt;
  }

  // Emit padded fragments for the main WMMA kernel.
  if (tid < 256) {
    int k = tid >> 4, n = tid & 15;
    ws[tid] = (k < KD && n < KD) ? Mbuf[cur][k][n] : 0.0f;
  }
  if (tid < 16) ws[256 + tid] = (tid < KD) ? vbuf[cur][tid] : 0.0f;
}

// ---------------------------------------------------------------------------
// Kernel 2 (hot): out = x @ M + v via V_WMMA_F32_16X16X4_F32.
// One wave32 per FULL 16-row tile; branch-free memory path.
//
// A 16x4 fp32 layout (ISA 7.12.2): lanes 0-15 hold M=lane, K={0,1};
//   lanes 16-31 hold M=lane-16, K={2,3}.
// B 4x16 layout: VGPR j, lanes 0-15 -> row K=j, lanes 16-31 -> row K=j+2.
// C/D 16x16 layout: VGPR p, lanes 0-15 -> row p, lanes 16-31 -> row p+8.
// ---------------------------------------------------------------------------
__global__ __launch_bounds__(256) void apply_affine_wmma(
    const float* __restrict__ x, const float* __restrict__ ws,
    float* __restrict__ out, int ntiles) {
  const int lane = threadIdx.x & 31;
  const int wid = (int)(blockIdx.x * (blockDim.x >> 5) + (threadIdx.x >> 5));
  if (wid >= ntiles) return;  // wave-uniform guard

  const int n  = lane & 15;         // column (B/C/D fragments)
  const int r  = lane & 15;         // row within tile (A fragment)
  const int kb = (lane >> 4) * 2;   // K sub-offset per half-wave
  const bool lo = lane < 16;

  // B fragments: ws is zero-padded 16x16, so rows k=10,11 read 0 -> no masks.
  v2f bf0, bf1, bf2;
  bf0.x = ws[(0 + kb) * 16 + n];  bf0.y = ws[(1 + kb) * 16 + n];
  bf1.x = ws[(4 + kb) * 16 + n];  bf1.y = ws[(5 + kb) * 16 + n];
  bf2.x = ws[(8 + kb) * 16 + n];  bf2.y = ws[(9 + kb) * 16 + n];

  // C init: effective bias broadcast down every row of the tile.
  const float bias = ws[256 + n];
  v8f c;
#pragma unroll
  for (int p = 0; p < 8; ++p) c[p] = bias;

  // A fragments: three unconditional 8B-aligned b64 loads per lane.
  // Chunk 2: upper half-wave (k=10,11) re-loads k=8,9 (always in-bounds),
  // then selects 0 via cndmask -- no exec manipulation.
  const float* xr = x + (size_t)wid * (16 * KD) + (size_t)r * KD;
  v2f a0 = __builtin_nontemporal_load((const v2f*)(xr + kb));      // k=kb,kb+1
  v2f a1 = __builtin_nontemporal_load((const v2f*)(xr + 4 + kb));  // k=4+kb,5+kb
  v2f a2r = __builtin_nontemporal_load((const v2f*)(xr + 8));      // k=8,9
  v2f a2;
  a2.x = lo ? a2r.x : 0.0f;
  a2.y = lo ? a2r.y : 0.0f;

  // D = A*B + C, chained over the 3 K-chunks (K = 0..11, rows 10,11 zero).
  c = __builtin_amdgcn_wmma_f32_16x16x4_f32(false, a0, false, bf0,
                                            (short)0, c, false, false);
  c = __builtin_amdgcn_wmma_f32_16x16x4_f32(false, a1, false, bf1,
                                            (short)0, c, false, false);
  c = __builtin_amdgcn_wmma_f32_16x16x4_f32(false, a2, false, bf2,
                                            (short)0, c, false, false);

  // Stores: single divergent region (n<10), 8 b32 stores off one base VGPR.
  if (n < KD) {
    float* orow = out + (size_t)wid * (16 * KD) + (size_t)(lo ? 0 : 8 * KD) + n;
#pragma unroll
    for (int p = 0; p < 8; ++p)
      __builtin_nontemporal_store(c[p], orow + p * KD);
  }
}

// ---------------------------------------------------------------------------
// Tail kernel: plain per-row affine for the <16 leftover rows (not launched
// for the reference shapes, where nrows % 16 == 0).
// ---------------------------------------------------------------------------
__global__ __launch_bounds__(32) void apply_affine_tail(
    const float* __restrict__ x, const float* __restrict__ ws,
    float* __restrict__ out, int startrow, int nrows) {
  const int row = startrow + (int)threadIdx.x;
  if (row >= nrows) return;
  float xi[KD];
#pragma unroll
  for (int k = 0; k < KD; ++k) xi[k] = x[(size_t)row * KD + k];
#pragma unroll
  for (int nn = 0; nn < KD; ++nn) {
    float acc = ws[256 + nn];
#pragma unroll
    for (int k = 0; k < KD; ++k) acc += xi[k] * ws[k * 16 + nn];
    out[(size_t)row * KD + nn] = acc;
  }
}

// ---------------------------------------------------------------------------
extern "C" void kernel_launch(void* const* d_in, const int* in_sizes, int n_in,
                              void* d_out, int out_size, void* d_ws, size_t ws_size,
                              hipStream_t stream) {
  const float* x = (const float*)d_in[0];
  const float* W = (const float*)d_in[1];
  const float* b = (const float*)d_in[2];
  const int* steps = (const int*)d_in[3];
  float* ws = (float*)d_ws;
  float* out = (float*)d_out;

  const int nrows = in_sizes[0] / KD;      // 1,000,000
  const int ntiles_full = nrows / 16;      // 62,500 full wave-tiles
  const int tail = nrows - ntiles_full * 16;

  precompute_affine<<<1, 256, 0, stream>>>(W, b, steps, ws);

  if (ntiles_full > 0) {
    const int waves_per_block = 256 / 32;  // wave32: 8 waves/block
    const int blocks = (ntiles_full + waves_per_block - 1) / waves_per_block;
    apply_affine_wmma<<<blocks, 256, 0, stream>>>(x, ws, out, ntiles_full);
  }
  if (tail > 0) {
    apply_affine_tail<<<1, 32, 0, stream>>>(x, ws, out, ntiles_full * 16, nrows);
  }
}